// GATConvolution_81140522156080
// MI455X (gfx1250) — compile-verified
//
#include <hip/hip_runtime.h>
#include <math.h>

#define F_IN 165
#define HID  64
#define NCLS 2
#define NEG_SLOPE 0.2f

typedef float v2f __attribute__((ext_vector_type(2)));
typedef float v8f __attribute__((ext_vector_type(8)));

// ---------- helpers: order-preserving float <-> uint for atomicMax ----------
__device__ __forceinline__ unsigned fkey(float f) {
    unsigned u = __float_as_uint(f);
    return (u & 0x80000000u) ? ~u : (u | 0x80000000u);
}
__device__ __forceinline__ float funkey(unsigned k) {
    return (k & 0x80000000u) ? __uint_as_float(k ^ 0x80000000u)
                             : __uint_as_float(~k);
}

// ---------- generic fills ----------
__global__ void fill_f32(float* p, float v, long n) {
    long i = (long)blockIdx.x * blockDim.x + threadIdx.x;
    if (i < n) p[i] = v;
}
__global__ void fill_u32(unsigned* p, unsigned v, long n) {
    long i = (long)blockIdx.x * blockDim.x + threadIdx.x;
    if (i < n) p[i] = v;
}

// ---------- layer-1 GEMM: H[N,64] = X[N,165] @ W[165,64] via f32 WMMA ----------
// Block = 256 threads = 8 wave32 waves; each wave owns one 16x16 C tile,
// arranged 2 (M) x 4 (N): block covers 32 rows x all 64 cols.
// K split: 41 branch-free steps (K=0..163) + 1 value-masked tail (K=164).
__global__ __launch_bounds__(256)
void gat_gemm1(const float* __restrict__ X, const float* __restrict__ W,
               float* __restrict__ H, int N) {
    __shared__ float sW[F_IN * HID];          // 42.24 KB of 320 KB LDS
    for (int i = threadIdx.x; i < F_IN * HID; i += 256) sW[i] = W[i];
    __syncthreads();

    const int lane = threadIdx.x & 31;
    const int half = lane >> 4;               // 0: K+0..1, 1: K+2..3
    const int l15  = lane & 15;
    const int wave = threadIdx.x >> 5;
    const int mBase = blockIdx.x * 32 + (wave >> 2) * 16;
    const int cBase = (wave & 3) * 16;
    const int col   = cBase + l15;

    int rowA = mBase + l15;
    if (rowA > N - 1) rowA = N - 1;           // branchless clamp (N%32==0 normally)
    const float* __restrict__ Arow = X + (long)rowA * F_IN;
    const int kh = 2 * half;

    // pull the row's cachelines toward the WGP (speculative)
    __builtin_prefetch(Arow, 0, 1);
    __builtin_prefetch(Arow + 96, 0, 1);

    v8f acc = {0.f, 0.f, 0.f, 0.f, 0.f, 0.f, 0.f, 0.f};

#pragma unroll
    for (int k0 = 0; k0 + 4 <= F_IN - 1; k0 += 4) {   // k0 = 0..160, all in bounds
        v2f a, b;
        a.x = Arow[k0 + kh];
        a.y = Arow[k0 + kh + 1];
        b.x = sW[(k0 + kh) * HID + col];
        b.y = sW[(k0 + kh + 1) * HID + col];
        acc = __builtin_amdgcn_wmma_f32_16x16x4_f32(
            false, a, false, b, (short)0, acc, false, false);
    }
    {   // tail: only K=164 is real; pad by value-select, never by EXEC
        const float ta = Arow[F_IN - 1];
        const float tb = sW[(F_IN - 1) * HID + col];
        v2f a, b;
        a.x = (half == 0) ? ta : 0.f;  a.y = 0.f;
        b.x = (half == 0) ? tb : 0.f;  b.y = 0.f;
        acc = __builtin_amdgcn_wmma_f32_16x16x4_f32(
            false, a, false, b, (short)0, acc, false, false);
    }

    const int rBase = mBase + 8 * half;       // VGPR r holds row rBase + r
    if (mBase + 31 < N) {                     // uniform fast path
#pragma unroll
        for (int r = 0; r < 8; ++r)
            H[(long)(rBase + r) * HID + col] = acc[r];
    } else {
#pragma unroll
        for (int r = 0; r < 8; ++r)
            if (rBase + r < N) H[(long)(rBase + r) * HID + col] = acc[r];
    }
}

// ---------- per-node attention scores: es = h.a_src, ed = h.a_dst ----------
// one wave32 per node, lane-parallel over the 64 features
__global__ __launch_bounds__(256)
void gat_scores(const float* __restrict__ H, const float* __restrict__ as,
                const float* __restrict__ ad, float* es, float* ed, int N) {
    const int node = (blockIdx.x * 256 + threadIdx.x) >> 5;
    const int lane = threadIdx.x & 31;
    if (node >= N) return;
    const float h0 = H[(long)node * HID + lane];
    const float h1 = H[(long)node * HID + 32 + lane];
    float ps = h0 * as[lane] + h1 * as[lane + 32];
    float pd = h0 * ad[lane] + h1 * ad[lane + 32];
#pragma unroll
    for (int off = 16; off; off >>= 1) {
        ps += __shfl_xor(ps, off, 32);
        pd += __shfl_xor(pd, off, 32);
    }
    if (lane == 0) { es[node] = ps; ed[node] = pd; }
}

// ---------- edge pass 1: segment max of leaky_relu(es[s]+ed[d]) over dst ----------
__global__ void gat_edge_max(const int* __restrict__ ei, const float* __restrict__ es,
                             const float* __restrict__ ed, unsigned* __restrict__ emax,
                             int E, int Etot) {
    const int idx = blockIdx.x * 256 + threadIdx.x;
    if (idx >= Etot) return;
    int s, d;
    if (idx < E) { s = ei[idx]; d = ei[E + idx]; } else { s = d = idx - E; }
    float e = es[s] + ed[d];
    e = (e >= 0.f) ? e : NEG_SLOPE * e;
    atomicMax(&emax[d], fkey(e));
}

// ---------- edge pass 2: ee = exp(e - emax[d]); denom[d] += ee ----------
__global__ void gat_edge_exp(const int* __restrict__ ei, const float* __restrict__ es,
                             const float* __restrict__ ed, const unsigned* __restrict__ emax,
                             float* __restrict__ ee, float* __restrict__ denom,
                             int E, int Etot) {
    const int idx = blockIdx.x * 256 + threadIdx.x;
    if (idx >= Etot) return;
    int s, d;
    if (idx < E) { s = ei[idx]; d = ei[E + idx]; } else { s = d = idx - E; }
    float e = es[s] + ed[d];
    e = (e >= 0.f) ? e : NEG_SLOPE * e;
    const float v = expf(e - funkey(emax[d]));
    ee[idx] = v;
    atomicAdd(&denom[d], v);
}

// ---------- edge pass 3 (layer 1): acc[d,:] += alpha * H[s,:], 64 feats ----------
// one wave per edge; each lane handles features lane and lane+32 (coalesced)
__global__ void gat_edge_aggr64(const int* __restrict__ ei, const float* __restrict__ ee,
                                const float* __restrict__ denom, const float* __restrict__ H,
                                float* __restrict__ acc, int E, int Etot) {
    const long gid  = (long)blockIdx.x * 256 + threadIdx.x;
    const int  edge = (int)(gid >> 5);
    if (edge >= Etot) return;
    const int lane = (int)(gid & 31);
    int s, d;
    if (edge < E) { s = ei[edge]; d = ei[E + edge]; } else { s = d = edge - E; }
    const float alpha = ee[edge] / denom[d];
    atomicAdd(&acc[(long)d * HID + lane],      alpha * H[(long)s * HID + lane]);
    atomicAdd(&acc[(long)d * HID + 32 + lane], alpha * H[(long)s * HID + 32 + lane]);
}

// ---------- layer-1 finalize: in-place relu(acc + b1) ----------
__global__ void gat_final1(float* __restrict__ acc, const float* __restrict__ b, long total) {
    const long i = (long)blockIdx.x * 256 + threadIdx.x;
    if (i >= total) return;
    const float v = acc[i] + b[i & (HID - 1)];
    acc[i] = v > 0.f ? v : 0.f;
}

// ---------- layer 2 per-node: h2 = x2 @ W2 (64x2), scores es2/ed2 ----------
__global__ void gat_layer2_node(const float* __restrict__ X2, const float* __restrict__ W2,
                                const float* __restrict__ a2s, const float* __restrict__ a2d,
                                float* __restrict__ H2, float* __restrict__ es2,
                                float* __restrict__ ed2, int N) {
    const int i = blockIdx.x * 256 + threadIdx.x;
    if (i >= N) return;
    float h0 = 0.f, h1 = 0.f;
#pragma unroll 8
    for (int k = 0; k < HID; ++k) {
        const float x = X2[(long)i * HID + k];
        h0 += x * W2[k * NCLS + 0];
        h1 += x * W2[k * NCLS + 1];
    }
    H2[i * 2 + 0] = h0;
    H2[i * 2 + 1] = h1;
    es2[i] = h0 * a2s[0] + h1 * a2s[1];
    ed2[i] = h0 * a2d[0] + h1 * a2d[1];
}

// ---------- edge pass 3 (layer 2): acc2[d,:2] += alpha * H2[s,:2] ----------
__global__ void gat_edge_aggr2(const int* __restrict__ ei, const float* __restrict__ ee,
                               const float* __restrict__ denom, const float* __restrict__ H2,
                               float* __restrict__ acc2, int E, int Etot) {
    const int idx = blockIdx.x * 256 + threadIdx.x;
    if (idx >= Etot) return;
    int s, d;
    if (idx < E) { s = ei[idx]; d = ei[E + idx]; } else { s = d = idx - E; }
    const float alpha = ee[idx] / denom[d];
    atomicAdd(&acc2[d * 2 + 0], alpha * H2[s * 2 + 0]);
    atomicAdd(&acc2[d * 2 + 1], alpha * H2[s * 2 + 1]);
}

// ---------- layer-2 finalize: out = acc2 + b2 ----------
__global__ void gat_final2(const float* __restrict__ acc2, const float* __restrict__ b2,
                           float* __restrict__ out, long total) {
    const long i = (long)blockIdx.x * 256 + threadIdx.x;
    if (i >= total) return;
    out[i] = acc2[i] + b2[i & (NCLS - 1)];
}

// ---------- copy edge_index (int32 bits) into d_out tail ----------
__global__ void copy_edge_index(const int* __restrict__ ei, int* __restrict__ dst, long n) {
    const long i = (long)blockIdx.x * 256 + threadIdx.x;
    if (i < n) dst[i] = ei[i];
}

extern "C" void kernel_launch(void* const* d_in, const int* in_sizes, int n_in,
                              void* d_out, int out_size, void* d_ws, size_t ws_size,
                              hipStream_t stream) {
    const float* x   = (const float*)d_in[0];
    const int*   ei  = (const int*)  d_in[1];
    const float* W1  = (const float*)d_in[2];
    const float* a1s = (const float*)d_in[3];
    const float* a1d = (const float*)d_in[4];
    const float* b1  = (const float*)d_in[5];
    const float* W2  = (const float*)d_in[6];
    const float* a2s = (const float*)d_in[7];
    const float* a2d = (const float*)d_in[8];
    const float* b2  = (const float*)d_in[9];

    const int N    = in_sizes[0] / F_IN;
    const int E    = in_sizes[1] / 2;
    const int Etot = E + N;          // with self loops

    // ---- workspace layout (floats) ----
    float*    ws     = (float*)d_ws;
    float*    h1     = ws;                       // N*HID
    float*    acc1   = h1   + (long)N * HID;     // N*HID (becomes x2 after relu)
    float*    es1    = acc1 + (long)N * HID;     // N
    float*    ed1    = es1  + N;                 // N
    float*    den1   = ed1  + N;                 // N
    unsigned* emax1  = (unsigned*)(den1 + N);    // N
    float*    den2   = (float*)(emax1 + N);      // N
    unsigned* emax2  = (unsigned*)(den2 + N);    // N
    float*    es2    = (float*)(emax2 + N);      // N
    float*    ed2    = es2  + N;                 // N
    float*    h2     = ed2  + N;                 // N*NCLS
    float*    acc2   = h2   + (long)N * NCLS;    // N*NCLS
    float*    eebuf  = acc2 + (long)N * NCLS;    // Etot

    float* out = (float*)d_out;

    const int T = 256;
    auto blk = [](long n, int t) { return (int)((n + t - 1) / t); };

    // ---- init accumulators ----
    fill_f32<<<blk((long)N * HID, T), T, 0, stream>>>(acc1, 0.f, (long)N * HID);
    fill_f32<<<blk(N, T), T, 0, stream>>>(den1, 0.f, N);
    fill_f32<<<blk(N, T), T, 0, stream>>>(den2, 0.f, N);
    fill_f32<<<blk((long)N * NCLS, T), T, 0, stream>>>(acc2, 0.f, (long)N * NCLS);
    fill_u32<<<blk(N, T), T, 0, stream>>>(emax1, 0u, N);  // key(x) > 0 for all finite x
    fill_u32<<<blk(N, T), T, 0, stream>>>(emax2, 0u, N);

    // ---- layer 1 ----
    gat_gemm1<<<blk(N, 32), T, 0, stream>>>(x, W1, h1, N);
    gat_scores<<<blk((long)N * 32, T), T, 0, stream>>>(h1, a1s, a1d, es1, ed1, N);
    gat_edge_max<<<blk(Etot, T), T, 0, stream>>>(ei, es1, ed1, emax1, E, Etot);
    gat_edge_exp<<<blk(Etot, T), T, 0, stream>>>(ei, es1, ed1, emax1, eebuf, den1, E, Etot);
    gat_edge_aggr64<<<blk((long)Etot * 32, T), T, 0, stream>>>(ei, eebuf, den1, h1, acc1, E, Etot);
    gat_final1<<<blk((long)N * HID, T), T, 0, stream>>>(acc1, b1, (long)N * HID);

    // ---- layer 2 ----
    gat_layer2_node<<<blk(N, T), T, 0, stream>>>(acc1, W2, a2s, a2d, h2, es2, ed2, N);
    gat_edge_max<<<blk(Etot, T), T, 0, stream>>>(ei, es2, ed2, emax2, E, Etot);
    gat_edge_exp<<<blk(Etot, T), T, 0, stream>>>(ei, es2, ed2, emax2, eebuf, den2, E, Etot);
    gat_edge_aggr2<<<blk(Etot, T), T, 0, stream>>>(ei, eebuf, den2, h2, acc2, E, Etot);
    gat_final2<<<blk((long)N * NCLS, T), T, 0, stream>>>(acc2, b2, out, (long)N * NCLS);

    // ---- second tuple element: edge_index bits appended to d_out ----
    copy_edge_index<<<blk((long)2 * E, T), T, 0, stream>>>(
        ei, (int*)d_out + (long)N * NCLS, (long)2 * E);
}